// DistanceWeightedMessagePassing_30477087933115
// MI455X (gfx1250) — compile-verified
//
#include <hip/hip_runtime.h>
#include <hip/hip_bf16.h>
#include <cstdint>

typedef __attribute__((ext_vector_type(2))) float v2f;
typedef __attribute__((ext_vector_type(8))) float v8f;

#define F_OUT      64
#define K_NBR      32
#define OUT_STRIDE 448   // 2*64*3 + 64

// ---------------------------------------------------------------------------
// Dense + bias + ReLU via V_WMMA_F32_16X16X4_F32 (fp32-exact, wave32).
// Block = 256 threads = 8 waves, arranged 2 (row tiles) x 4 (col tiles):
// 32 vertex-rows x 64 output channels per block.
// A (16x4 f32): lanes 0-15 hold M=0..15 / K={k0,k0+1}; lanes 16-31 K={k0+2,k0+3}.
// B (4x16 f32): lanes hold N=lane%16; half-wave selects K pair (same kOff).
// C/D (16x16 f32): VGPR r -> M = r + 8*(lane>=16), N = lane%16.
// ---------------------------------------------------------------------------
__global__ __launch_bounds__(256)
void dense_relu_wmma(const float* __restrict__ in, int inStride, int Cin,
                     const float* __restrict__ W, const float* __restrict__ bias,
                     float* __restrict__ outF, int V)
{
    const int wave = threadIdx.x >> 5;
    const int lane = threadIdx.x & 31;
    const int m    = lane & 15;
    const int half = lane >> 4;      // 0 or 1
    const int kOff = half * 2;       // K offset within the 4-wide K tile

    const int rowTile = blockIdx.x * 32 + (wave >> 2) * 16;
    const int colTile = (wave & 3) * 16;

    int rowLd = rowTile + m;
    if (rowLd > V - 1) rowLd = V - 1;          // clamp loads, keep EXEC all-1s
    const float* inRow = in + (size_t)rowLd * inStride;
    const float* Wc    = W + colTile + m;      // column of W, stride F_OUT per K

    v8f acc = {};
    for (int k0 = 0; k0 < Cin; k0 += 4) {
        const float2 av = *(const float2*)(inRow + k0 + kOff);   // global_load_b64
        v2f a, b;
        a.x = av.x;
        a.y = av.y;
        b.x = Wc[(size_t)(k0 + kOff)     * F_OUT];
        b.y = Wc[(size_t)(k0 + kOff + 1) * F_OUT];
        acc = __builtin_amdgcn_wmma_f32_16x16x4_f32(
                  false, a, false, b, (short)0, acc, false, false);
    }

    const int   n     = colTile + m;
    const float bn    = bias[n];
    const int   mBase = rowTile + half * 8;
#pragma unroll
    for (int r = 0; r < 8; ++r) {
        const int rr = mBase + r;
        if (rr < V) {
            outF[(size_t)rr * F_OUT + n] = fmaxf(acc[r] + bn, 0.0f);
        }
    }
}

// ---------------------------------------------------------------------------
// KNN aggregation, fused with "agg - tile(prev)" and strided write into d_out.
// One wave per vertex; lane l owns channels {2l, 2l+1}. Each lane loads its
// own neighbor idx + weight once; the k-loop broadcasts them via wave32
// shuffles and does one float2 (8B) gather per lane -> one full 64-wide
// feature row per wave per neighbor. feats table (25.6 MB) lives in L2.
// ---------------------------------------------------------------------------
__global__ __launch_bounds__(256)
void knn_aggregate(const float* __restrict__ feats, const int* __restrict__ nidx,
                   const float* __restrict__ d2, float* __restrict__ out,
                   int colBase, int V)
{
    const int wave = threadIdx.x >> 5;
    const int lane = threadIdx.x & 31;
    const int v    = blockIdx.x * 8 + wave;
    if (v >= V) return;

    const int   myN = nidx[(size_t)v * K_NBR + lane];
    const float myW = __expf(-10.0f * d2[(size_t)v * K_NBR + lane]);

    // Warm this lane's neighbor row (256B = two prefetch points).
    const float* myRow = feats + (size_t)myN * F_OUT;
    __builtin_prefetch(myRow,      0, 0);   // global_prefetch_b8
    __builtin_prefetch(myRow + 32, 0, 0);

    const int c = lane * 2;
    float sx = 0.0f, sy = 0.0f;
    float mx = -__builtin_huge_valf(), my = -__builtin_huge_valf();

#pragma unroll 4
    for (int k = 0; k < K_NBR; ++k) {
        const int   n = __shfl(myN, k);
        const float w = __shfl(myW, k);
        const float2 f = *(const float2*)(feats + (size_t)n * F_OUT + c);
        const float wx = f.x * w;
        const float wy = f.y * w;
        sx += wx; sy += wy;
        mx = fmaxf(mx, wx);
        my = fmaxf(my, wy);
    }

    const float2 prev = *(const float2*)(feats + (size_t)v * F_OUT + c);
    const float  invK = 1.0f / (float)K_NBR;
    float* o = out + (size_t)v * OUT_STRIDE + colBase;
    *(float2*)(o + c)         = make_float2(sx * invK - prev.x, sy * invK - prev.y);
    *(float2*)(o + F_OUT + c) = make_float2(mx - prev.x,        my - prev.y);
}

// ---------------------------------------------------------------------------
// Pass-through copy of x into out[:, 384:448], 16B vectorized.
// ---------------------------------------------------------------------------
__global__ __launch_bounds__(256)
void copy_x_kernel(const float* __restrict__ x, float* __restrict__ out, int V)
{
    const int t = blockIdx.x * blockDim.x + threadIdx.x;
    if (t >= V * 16) return;
    const int v = t >> 4;
    const int c = (t & 15) * 4;
    *(float4*)(out + (size_t)v * OUT_STRIDE + 384 + c) =
        *(const float4*)(x + (size_t)v * 64 + c);
}

extern "C" void kernel_launch(void* const* d_in, const int* in_sizes, int n_in,
                              void* d_out, int out_size, void* d_ws, size_t ws_size,
                              hipStream_t stream)
{
    const float* x   = (const float*)d_in[0];
    const int*   idx = (const int*)  d_in[1];
    const float* d2  = (const float*)d_in[2];
    const float* W0  = (const float*)d_in[3];
    const float* b0  = (const float*)d_in[4];
    const float* W1  = (const float*)d_in[5];
    const float* b1  = (const float*)d_in[6];
    const float* W2  = (const float*)d_in[7];
    const float* b2  = (const float*)d_in[8];
    float* out = (float*)d_out;

    const int V = in_sizes[0] / 64;

    // Workspace: single V x 64 fp32 feats buffer (25.6 MB), resident in L2.
    float* featsBuf = (float*)d_ws;

    const dim3 denseGrid((V + 31) / 32), blk256(256);
    const dim3 knnGrid((V + 7) / 8);
    const dim3 cpGrid((V * 16 + 255) / 256);

    // Layer 0: x [V,64] -> feats -> out[:, 0:128)
    dense_relu_wmma<<<denseGrid, blk256, 0, stream>>>(x, 64, 64, W0, b0, featsBuf, V);
    knn_aggregate  <<<knnGrid,   blk256, 0, stream>>>(featsBuf, idx, d2, out, 0, V);

    // Layer 1: out[:, 0:128) -> feats -> out[:, 128:256)
    dense_relu_wmma<<<denseGrid, blk256, 0, stream>>>(out, OUT_STRIDE, 128, W1, b1, featsBuf, V);
    knn_aggregate  <<<knnGrid,   blk256, 0, stream>>>(featsBuf, idx, d2, out, 128, V);

    // Layer 2: out[:, 128:256) -> feats -> out[:, 256:384)
    dense_relu_wmma<<<denseGrid, blk256, 0, stream>>>(out + 128, OUT_STRIDE, 128, W2, b2, featsBuf, V);
    knn_aggregate  <<<knnGrid,   blk256, 0, stream>>>(featsBuf, idx, d2, out, 256, V);

    // x pass-through into out[:, 384:448)
    copy_x_kernel<<<cpGrid, blk256, 0, stream>>>(x, out, V);
}